// GCNStack_57166014710074
// MI455X (gfx1250) — compile-verified
//
#include <hip/hip_runtime.h>
#include <math.h>

typedef __attribute__((ext_vector_type(2))) float v2f;
typedef __attribute__((ext_vector_type(4))) float v4f;
typedef __attribute__((ext_vector_type(8))) float v8f;

#define B_DIM 64
#define T_DIM 256
#define E_DIM 64
#define R_DIM 4
#define S_DIM 8

// ---------------------------------------------------------------------------
// Gram matrix via f32 WMMA: out[i,j] = scale * sum_k X[i,k]*X[j,k]
// X: [64, K] row-major, out: [64, 64].
// grid = 16 blocks (4x4 tiles of 16x16), block = 256 threads (8 waves, K-split)
//
// K-permuted operand staging: for each unrolled group of 16 K-values starting
// at k0, lane-half h (= lane>>4) supplies the contiguous range
// [k0 + 8h, k0 + 8h + 8) across the group's 4 WMMAs (2 values per WMMA slot).
// A and B use the same kappa(h,v) mapping, so the Gram sum over K is exact.
// Per lane per group: 2 contiguous b128 loads for A, 2 for B.
//
// C/D 16x16 f32 layout: VGPR v, lane L -> M = v + 8*(L>>4), N = L&15
// ---------------------------------------------------------------------------
template <int K>
__global__ void __launch_bounds__(256)
gram_wmma_kernel(const float* __restrict__ X, float* __restrict__ out,
                 float scale) {
  const int tile = blockIdx.x;
  const int m0   = (tile >> 2) << 4;
  const int n0   = (tile & 3) << 4;
  const int wave = __builtin_amdgcn_readfirstlane((int)threadIdx.x) >> 5; // scalar
  const int lane = threadIdx.x & 31;
  const int row  = lane & 15;
  const int h8   = (lane >> 4) << 3;   // 0 or 8: contiguous K sub-range base

  const float* arow = X + (size_t)(m0 + row) * K + h8;
  const float* brow = X + (size_t)(n0 + row) * K + h8;

  constexpr int kslice = K >> 3;       // per-wave K slice (multiple of 16)
  const int kbeg = wave * kslice;
  const int kend = kbeg + kslice;

  v8f c = {0.f, 0.f, 0.f, 0.f, 0.f, 0.f, 0.f, 0.f};
  for (int k = kbeg; k < kend; k += 16) {
    const v4f a0 = *(const v4f*)(arow + k);
    const v4f a1 = *(const v4f*)(arow + k + 4);
    const v4f b0 = *(const v4f*)(brow + k);
    const v4f b1 = *(const v4f*)(brow + k + 4);

    v2f av, bv;
    av[0] = a0[0]; av[1] = a0[1]; bv[0] = b0[0]; bv[1] = b0[1];
    c = __builtin_amdgcn_wmma_f32_16x16x4_f32(false, av, false, bv, (short)0, c, false, false);
    av[0] = a0[2]; av[1] = a0[3]; bv[0] = b0[2]; bv[1] = b0[3];
    c = __builtin_amdgcn_wmma_f32_16x16x4_f32(false, av, false, bv, (short)0, c, false, false);
    av[0] = a1[0]; av[1] = a1[1]; bv[0] = b1[0]; bv[1] = b1[1];
    c = __builtin_amdgcn_wmma_f32_16x16x4_f32(false, av, false, bv, (short)0, c, false, false);
    av[0] = a1[2]; av[1] = a1[3]; bv[0] = b1[2]; bv[1] = b1[3];
    c = __builtin_amdgcn_wmma_f32_16x16x4_f32(false, av, false, bv, (short)0, c, false, false);
  }

  __shared__ float red[8][256];
  #pragma unroll
  for (int v = 0; v < 8; ++v) {
    const int M = v + ((lane >> 4) << 3);
    const int N = lane & 15;
    red[wave][M * 16 + N] = c[v];
  }
  __syncthreads();

  float s = 0.f;
  #pragma unroll
  for (int w = 0; w < 8; ++w) s += red[w][threadIdx.x];
  const int M = threadIdx.x >> 4;
  const int N = threadIdx.x & 15;
  out[(size_t)(m0 + M) * B_DIM + (n0 + N)] = s * scale;
}

// ---------------------------------------------------------------------------
// Row-wise sparsemax on a [64, 64] matrix. grid = 64 rows, block = 64 threads.
// Rank-based O(n^2) formulation (no sort needed).
// ---------------------------------------------------------------------------
__global__ void __launch_bounds__(64)
sparsemax_kernel(const float* __restrict__ in, float* __restrict__ out) {
  __shared__ float z[64];
  __shared__ int   rank[64];
  __shared__ int   kcount;
  __shared__ float tau_s;

  const int i = threadIdx.x;
  const float zi = in[(size_t)blockIdx.x * 64 + i];
  z[i] = zi;
  if (i == 0) kcount = 0;
  __syncthreads();

  // descending rank with index tie-break -> permutation of 1..64
  int r = 1;
  for (int j = 0; j < 64; ++j) {
    const float zj = z[j];
    if (zj > zi || (zj == zi && j < i)) ++r;
  }
  rank[i] = r;
  __syncthreads();

  // S = cumulative sum of sorted values up to this rank
  float S = 0.f;
  for (int j = 0; j < 64; ++j)
    if (rank[j] <= r) S += z[j];

  // support condition at sorted position r; k = number of true conditions
  const bool support = (1.0f + (float)r * zi) > S;
  if (support) atomicAdd(&kcount, 1);
  __syncthreads();
  if (r == kcount) tau_s = (S - 1.0f) / (float)r;  // exactly one thread
  __syncthreads();

  out[(size_t)blockIdx.x * 64 + i] = fmaxf(zi - tau_s, 0.0f);
}

// ---------------------------------------------------------------------------
// Layer-0 aggregation + matmul:  x [64,256,64], adj row -> y [64,256,4]
// agg[B,t,e] = sum_b max_{b' in [b-1,b+2] clipped} adj[B,b'] * x[b',t,e]
// y[B,t,r]   = relu(sum_e agg[B,t,e] * W[e,r])
// grid = (64 t-chunks, 64 B), block = 256 (4 t x 64 e)
// ---------------------------------------------------------------------------
__global__ void __launch_bounds__(256)
agg_mm_e64_kernel(const float* __restrict__ x, const float* __restrict__ adj,
                  const float* __restrict__ W, float* __restrict__ y) {
  const int B   = blockIdx.y;
  const int t0  = blockIdx.x * 4;
  const int tid = threadIdx.x;
  const int tl  = tid >> 6;     // 0..3
  const int e   = tid & 63;
  const int t   = t0 + tl;

  __shared__ float a_s[64];
  __shared__ float agg_s[4][64];
  if (tid < 64) a_s[tid] = adj[(size_t)B * 64 + tid];
  __syncthreads();

  const float* xcol = x + (size_t)t * E_DIM + e;   // x[b',t,e]
  const size_t strideB = (size_t)T_DIM * E_DIM;

  // streaming windowed max-sum with a 4-slot ring: window for b is [b-1, b+2]
  float v[4];
  v[0] = a_s[0] * xcol[0];
  v[1] = a_s[1] * xcol[strideB];
  float acc = 0.f;
  for (int b = 0; b < 64; ++b) {
    const int jn = b + 2;
    if (jn < 64) v[jn & 3] = a_s[jn] * xcol[(size_t)jn * strideB];
    float m = v[b & 3];
    if (b >= 1)     m = fmaxf(m, v[(b - 1) & 3]);
    if (b + 1 < 64) m = fmaxf(m, v[(b + 1) & 3]);
    if (jn < 64)    m = fmaxf(m, v[jn & 3]);
    acc += m;
  }
  agg_s[tl][e] = acc;
  __syncthreads();

  if (tid < 16) {
    const int tt = tid >> 2, r = tid & 3;
    float s = 0.f;
    for (int e2 = 0; e2 < 64; ++e2) s += agg_s[tt][e2] * W[e2 * R_DIM + r];
    y[((size_t)B * T_DIM + t0 + tt) * R_DIM + r] = fmaxf(s, 0.f);
  }
}

// ---------------------------------------------------------------------------
// Layer-1 aggregation + matmul: x [64,256,4], adj row -> y [64,256,4]
// grid = (4 t-chunks, 64 B), block = 256 (64 t x 4 r)
// ---------------------------------------------------------------------------
__global__ void __launch_bounds__(256)
agg_mm_e4_kernel(const float* __restrict__ x, const float* __restrict__ adj,
                 const float* __restrict__ W, float* __restrict__ y) {
  const int B   = blockIdx.y;
  const int t0  = blockIdx.x * 64;
  const int tid = threadIdx.x;
  const int tl  = tid >> 2;     // 0..63
  const int r   = tid & 3;
  const int t   = t0 + tl;

  __shared__ float a_s[64];
  __shared__ float agg_s[64][4];
  if (tid < 64) a_s[tid] = adj[(size_t)B * 64 + tid];
  __syncthreads();

  const float* xcol = x + (size_t)t * R_DIM + r;
  const size_t strideB = (size_t)T_DIM * R_DIM;

  float v[4];
  v[0] = a_s[0] * xcol[0];
  v[1] = a_s[1] * xcol[strideB];
  float acc = 0.f;
  for (int b = 0; b < 64; ++b) {
    const int jn = b + 2;
    if (jn < 64) v[jn & 3] = a_s[jn] * xcol[(size_t)jn * strideB];
    float m = v[b & 3];
    if (b >= 1)     m = fmaxf(m, v[(b - 1) & 3]);
    if (b + 1 < 64) m = fmaxf(m, v[(b + 1) & 3]);
    if (jn < 64)    m = fmaxf(m, v[jn & 3]);
    acc += m;
  }
  agg_s[tl][r] = acc;
  __syncthreads();

  float s = 0.f;
  #pragma unroll
  for (int r2 = 0; r2 < 4; ++r2) s += agg_s[tl][r2] * W[r2 * R_DIM + r];
  y[((size_t)B * T_DIM + t) * R_DIM + r] = fmaxf(s, 0.f);
}

// ---------------------------------------------------------------------------
// Head: mean over T, @ Wp [4,8], softmax. grid = 64 (B), block = 64
// ---------------------------------------------------------------------------
__global__ void __launch_bounds__(64)
head_kernel(const float* __restrict__ y, const float* __restrict__ Wp,
            float* __restrict__ out) {
  const int B   = blockIdx.x;
  const int tid = threadIdx.x;
  __shared__ float part[64][4];
  __shared__ float pooled[4];
  __shared__ float logits[8];

  float p[4] = {0.f, 0.f, 0.f, 0.f};
  for (int t = tid; t < T_DIM; t += 64) {
    #pragma unroll
    for (int r = 0; r < 4; ++r)
      p[r] += y[((size_t)B * T_DIM + t) * R_DIM + r];
  }
  #pragma unroll
  for (int r = 0; r < 4; ++r) part[tid][r] = p[r];
  __syncthreads();

  if (tid < 4) {
    float s = 0.f;
    for (int j = 0; j < 64; ++j) s += part[j][tid];
    pooled[tid] = s / (float)T_DIM;
  }
  __syncthreads();

  if (tid < 8) {
    float s = 0.f;
    #pragma unroll
    for (int r = 0; r < 4; ++r) s += pooled[r] * Wp[r * S_DIM + tid];
    logits[tid] = s;
  }
  __syncthreads();

  if (tid == 0) {
    float mx = logits[0];
    for (int j = 1; j < 8; ++j) mx = fmaxf(mx, logits[j]);
    float ex[8], den = 0.f;
    for (int j = 0; j < 8; ++j) { ex[j] = __expf(logits[j] - mx); den += ex[j]; }
    for (int j = 0; j < 8; ++j) out[(size_t)B * S_DIM + j] = ex[j] / den;
  }
}

// ---------------------------------------------------------------------------
extern "C" void kernel_launch(void* const* d_in, const int* in_sizes, int n_in,
                              void* d_out, int out_size, void* d_ws, size_t ws_size,
                              hipStream_t stream) {
  (void)in_sizes; (void)n_in; (void)out_size; (void)ws_size;
  const float* x  = (const float*)d_in[0];   // [64,256,64]
  const float* W0 = (const float*)d_in[1];   // [64,4]
  const float* W1 = (const float*)d_in[2];   // [4,4]
  const float* Wp = (const float*)d_in[3];   // [4,8]
  float* out = (float*)d_out;                // [64,8]

  float* f    = (float*)d_ws;
  float* adjA = f;                 // 4096  raw gram
  float* adjB = f + 4096;          // 4096  sparsemax output
  float* y0   = f + 8192;          // 65536 layer-0 output [64,256,4]
  float* y1   = f + 8192 + 65536;  // 65536 layer-1 output [64,256,4]

  // --- layer 0 ---
  gram_wmma_kernel<T_DIM * E_DIM><<<16, 256, 0, stream>>>(x, adjA, 1.0f / T_DIM);
  sparsemax_kernel<<<64, 64, 0, stream>>>(adjA, adjB);
  agg_mm_e64_kernel<<<dim3(64, 64), 256, 0, stream>>>(x, adjB, W0, y0);

  // --- layer 1 ---
  gram_wmma_kernel<T_DIM * R_DIM><<<16, 256, 0, stream>>>(y0, adjA, 1.0f / T_DIM);
  sparsemax_kernel<<<64, 64, 0, stream>>>(adjA, adjB);
  agg_mm_e4_kernel<<<dim3(4, 64), 256, 0, stream>>>(y0, adjB, W1, y1);

  // --- head ---
  head_kernel<<<64, 64, 0, stream>>>(y1, Wp, out);
}